// VirtualNode_76630806495690
// MI455X (gfx1250) — compile-verified
//
#include <hip/hip_runtime.h>

#define DIM 128
#define N_GRAPHS 1024

typedef __attribute__((ext_vector_type(2))) float v2f;
typedef __attribute__((ext_vector_type(8))) float v8f;

// ---------------------------------------------------------------------------
// Kernel A: per-graph mean pool (segment ids sorted -> binary search range),
// then X[g] = vn_h[g] + pool[g].  Deterministic: fixed-order LDS reduction.
// grid = N_GRAPHS blocks, 256 threads (8 row-streams x 32 float4 columns).
// ---------------------------------------------------------------------------
__global__ void seg_pool_kernel(const float* __restrict__ h,
                                const float* __restrict__ vn_h,
                                const int* __restrict__ seg,
                                float* __restrict__ X,
                                int n_nodes) {
    const int g = blockIdx.x;

    // lower_bound(seg, g) and lower_bound(seg, g+1): uniform across the block
    int lo = 0, hi = n_nodes;
    while (lo < hi) { int mid = (lo + hi) >> 1; if (seg[mid] <  g) lo = mid + 1; else hi = mid; }
    const int start = lo;
    hi = n_nodes;
    while (lo < hi) { int mid = (lo + hi) >> 1; if (seg[mid] <= g) lo = mid + 1; else hi = mid; }
    const int end = lo;

    const int quad   = threadIdx.x & 31;  // float4 column: dims [4q, 4q+3]
    const int rowoff = threadIdx.x >> 5;  // 0..7

    const float4* h4 = (const float4*)h;
    float4 acc = make_float4(0.f, 0.f, 0.f, 0.f);
    for (int r = start + rowoff; r < end; r += 8) {
        float4 v = h4[(size_t)r * 32 + quad];
        acc.x += v.x; acc.y += v.y; acc.z += v.z; acc.w += v.w;
    }

    __shared__ float4 red[8][32];
    red[rowoff][quad] = acc;
    __syncthreads();

    if (rowoff == 0) {
        float4 s = red[0][quad];
#pragma unroll
        for (int i = 1; i < 8; ++i) {
            float4 v = red[i][quad];
            s.x += v.x; s.y += v.y; s.z += v.z; s.w += v.w;
        }
        const float inv = 1.0f / fmaxf((float)(end - start), 1.0f);
        const float4* vn4 = (const float4*)vn_h;
        float4 x = vn4[(size_t)g * 32 + quad];
        x.x += s.x * inv; x.y += s.y * inv; x.z += s.z * inv; x.w += s.w * inv;
        ((float4*)X)[(size_t)g * 32 + quad] = x;
    }
}

// ---------------------------------------------------------------------------
// Kernel B: vn_new = vn_h + relu(X @ W + b) using V_WMMA_F32_16X16X4_F32.
// One 16x16 output tile per wave32; K=128 in steps of 4 (32 WMMAs per tile).
// A 16x4 f32 layout (ISA 7.12.2): lane%16 = M, lane/16 picks K-pair {0,1}/{2,3}
// held in the float2.  B 4x16 mirrored: lane%16 = N, lane/16 picks K-pair.
// C/D: VGPR v, lane l -> M = v + 8*(l/16), N = l%16.
// grid = 64 blocks x 256 threads = 512 waves = 512 tiles. No divergence.
// ---------------------------------------------------------------------------
__global__ void vn_gemm_wmma_kernel(const float* __restrict__ X,
                                    const float* __restrict__ W,
                                    const float* __restrict__ bias,
                                    const float* __restrict__ vn_h,
                                    float* __restrict__ vn_out) {
    const int wave = threadIdx.x >> 5;
    const int lane = threadIdx.x & 31;
    const int tile = blockIdx.x * 8 + wave;   // 0..511
    const int mt   = tile >> 3;               // 0..63  (M tile)
    const int nt   = tile & 7;                // 0..7   (N tile)

    const int mn   = lane & 15;               // M for A-frag, N for B-frag
    const int half = lane >> 4;               // 0/1 -> K-pair select

    const int rowA = mt * 16 + mn;            // A row in X
    const int colB = nt * 16 + mn;            // B col in W

    v8f c = {0.f, 0.f, 0.f, 0.f, 0.f, 0.f, 0.f, 0.f};

#pragma unroll
    for (int k0 = 0; k0 < DIM; k0 += 4) {
        const int ka = k0 + 2 * half;
        // A fragment: X[rowA][ka], X[rowA][ka+1]  (8B-aligned float2 load)
        v2f a = *(const v2f*)(X + (size_t)rowA * DIM + ka);
        // B fragment: W[ka][colB], W[ka+1][colB]
        v2f bfrag;
        bfrag.x = W[(size_t)(ka)     * DIM + colB];
        bfrag.y = W[(size_t)(ka + 1) * DIM + colB];
        c = __builtin_amdgcn_wmma_f32_16x16x4_f32(
                /*neg_a=*/false, a, /*neg_b=*/false, bfrag,
                /*c_mod=*/(short)0, c, /*reuse_a=*/false, /*reuse_b=*/false);
    }

    // Epilogue: + bias, ReLU, residual add of vn_h, store vn_new.
    const float bn = bias[colB];
#pragma unroll
    for (int v = 0; v < 8; ++v) {
        const int Mg = mt * 16 + v + 8 * half;
        float val = c[v] + bn;
        val = fmaxf(val, 0.0f);
        vn_out[(size_t)Mg * DIM + colB] = vn_h[(size_t)Mg * DIM + colB] + val;
    }
}

// ---------------------------------------------------------------------------
// Kernel C: h_new = h + vn_new[seg].  One float4 per thread; 32 consecutive
// lanes cover one row so seg[row] is wave-uniform (scalar load); vn_new is
// 512 KB -> L2 resident.  Pure streaming pass (1 GB of the 1.5 GB total).
// ---------------------------------------------------------------------------
__global__ void broadcast_kernel(const float* __restrict__ h,
                                 const float* __restrict__ vn_new,
                                 const int* __restrict__ seg,
                                 float* __restrict__ h_out,
                                 int n_nodes) {
    const size_t gid = (size_t)blockIdx.x * blockDim.x + threadIdx.x;
    const size_t total4 = (size_t)n_nodes * 32;
    if (gid >= total4) return;
    const int row = (int)(gid >> 5);
    const int q   = (int)(gid & 31);
    const int g   = seg[row];

    const float4 a = ((const float4*)h)[gid];
    const float4 v = ((const float4*)vn_new)[(size_t)g * 32 + q];
    float4 o;
    o.x = a.x + v.x; o.y = a.y + v.y; o.z = a.z + v.z; o.w = a.w + v.w;
    ((float4*)h_out)[gid] = o;
}

extern "C" void kernel_launch(void* const* d_in, const int* in_sizes, int n_in,
                              void* d_out, int out_size, void* d_ws, size_t ws_size,
                              hipStream_t stream) {
    const float* h    = (const float*)d_in[0];   // [N, 128]
    const float* vn_h = (const float*)d_in[1];   // [1024, 128]
    const int*   seg  = (const int*)d_in[2];     // [N] sorted
    const float* W    = (const float*)d_in[3];   // [128, 128]
    const float* bias = (const float*)d_in[4];   // [128]

    const int n_nodes = in_sizes[0] / DIM;

    float* X      = (float*)d_ws;                        // [1024, 128] staged vn_h + pool
    float* vn_out = (float*)d_out;                       // output 0: [1024, 128]
    float* h_out  = (float*)d_out + (size_t)N_GRAPHS * DIM;  // output 1: [N, 128]

    // Pass 1: segment mean-pool + add vn_h  (reads h once: 512 MB)
    seg_pool_kernel<<<N_GRAPHS, 256, 0, stream>>>(h, vn_h, seg, X, n_nodes);

    // Pass 2: tiny f32 WMMA GEMM + bias + ReLU + residual -> vn_new
    // tiles = (1024/16)*(128/16) = 512; 8 waves/block -> 64 blocks
    vn_gemm_wmma_kernel<<<64, 256, 0, stream>>>(X, W, bias, vn_h, vn_out);

    // Pass 3: broadcast vn_new back to nodes (reads h + writes h_new: 1 GB)
    const size_t total4 = (size_t)n_nodes * 32;
    const int blocks = (int)((total4 + 255) / 256);
    broadcast_kernel<<<blocks, 256, 0, stream>>>(h, vn_out, seg, h_out, n_nodes);
}